// HardRegionSensitiveLoss_2D_31576599560773
// MI455X (gfx1250) — compile-verified
//
#include <hip/hip_runtime.h>
#include <cstdint>

// HardRegionSensitiveLoss_2D for MI455X (gfx1250, wave32).
// preds: [16,4,512,512] f32, labels: [16,512,512] i32, out: scalar f32.
// patch=16, stride=8 -> 63x63 patches per image; mean over N*C*L = 254016.

#define HW      512
#define NCH     4
#define PS      16
#define PSTRIDE 8
#define OW      63
#define OH      63
#define NBATCH  16

// ---- CDNA5 async global->LDS (tracked by ASYNCcnt) ------------------------
__device__ __forceinline__ uint32_t lds_off_of(const void* p) {
  // Flat pointers into the LDS aperture carry the wave-relative LDS byte
  // offset in their low 32 bits (CDNA5 ISA 10.2).
  return (uint32_t)(uintptr_t)p;
}

__device__ __forceinline__ void async_to_lds_b128(uint32_t lds_byte_off,
                                                  uint32_t global_byte_off,
                                                  const void* base) {
  // GVS mode: 64-bit SGPR base + 32-bit per-lane VGPR byte offset.
  asm volatile("global_load_async_to_lds_b128 %0, %1, %2"
               :: "v"(lds_byte_off), "v"(global_byte_off), "s"(base)
               : "memory");
}

__device__ __forceinline__ void wait_async_zero() {
  asm volatile("s_wait_asynccnt 0" ::: "memory");
}

// ---- DPP lane exchange: stays in the VALU pipe (no LDS / dscnt waits) -----
template <int CTRL>
__device__ __forceinline__ float dppmov(float x) {
  return __int_as_float(__builtin_amdgcn_update_dpp(
      0, __float_as_int(x), CTRL, 0xf, 0xf, true));
}

// wave32 butterfly for (min, sum, sum). Ascending order is required:
// row_half_mirror (i^7) == xor4 and row_mirror (i^15) == xor8 only because
// lanes are already uniform within 4- and 8-groups from the earlier steps.
__device__ __forceinline__ void wave_reduce3(float& mn, float& ct, float& ls) {
  mn = fminf(mn, dppmov<0xB1>(mn));   // quad_perm(1,0,3,2): xor1
  ct += dppmov<0xB1>(ct);
  ls += dppmov<0xB1>(ls);
  mn = fminf(mn, dppmov<0x4E>(mn));   // quad_perm(2,3,0,1): xor2
  ct += dppmov<0x4E>(ct);
  ls += dppmov<0x4E>(ls);
  mn = fminf(mn, dppmov<0x141>(mn));  // row_half_mirror: xor4
  ct += dppmov<0x141>(ct);
  ls += dppmov<0x141>(ls);
  mn = fminf(mn, dppmov<0x140>(mn));  // row_mirror: xor8
  ct += dppmov<0x140>(ct);
  ls += dppmov<0x140>(ls);
  mn = fminf(mn, __shfl_xor(mn, 16, 32));  // cross-half: one ds op per value
  ct += __shfl_xor(ct, 16, 32);
  ls += __shfl_xor(ls, 16, 32);
}

// ---- per-patch kernel: one 256-thread block per (n, py, px) ---------------
__global__ void __launch_bounds__(256)
patch_loss_kernel(const float* __restrict__ preds,
                  const int*   __restrict__ labels,
                  float*       __restrict__ partial) {
  __shared__ float s_pred[NCH * 256];   // [c][row*16+col]
  __shared__ int   s_lab[256];          // [row*16+col]
  __shared__ float s_red[8 * 12];       // 8 waves x (4ch x {min,cnt,lgs})
  __shared__ float s_fin[12];           // combined 4ch x {min,cnt,lgs}

  const int px = blockIdx.x;
  const int py = blockIdx.y;
  const int n  = blockIdx.z;
  const int t  = threadIdx.x;

  // Stage pred tile: 4ch x 16row x 16col = 256 b128 chunks (4 floats each).
  {
    const int c   = t >> 6;          // channel
    const int row = (t >> 2) & 15;   // row within patch
    const int j   = t & 3;           // 4-float chunk within row
    const uint32_t goff =
        (uint32_t)(((((n * NCH + c) * HW) + (py * PSTRIDE + row)) * HW
                    + (px * PSTRIDE + 4 * j)) * 4);
    async_to_lds_b128(lds_off_of(&s_pred[c * 256 + (t & 63) * 4]), goff, preds);
  }
  // Stage label tile: 16row x 16col ints = 64 b128 chunks.
  if (t < 64) {
    const int row = t >> 2;
    const int j   = t & 3;
    const uint32_t goff =
        (uint32_t)((((n * HW) + (py * PSTRIDE + row)) * HW
                    + (px * PSTRIDE + 4 * j)) * 4);
    async_to_lds_b128(lds_off_of(&s_lab[t * 4]), goff, labels);
  }
  wait_async_zero();
  __syncthreads();

  // Per-pixel masked stats. label is one-hot across channels, so only ONE
  // log per pixel is ever needed (the min path needs raw p, not log p).
  const int   lab  = s_lab[t];                       // 0..3
  const float plab = s_pred[lab * 256 + t];          // bank = t%64: conflict-free
  const float lp   = __logf(plab + 1e-8f);           // single v_log_f32 / pixel

  float mn[NCH], cnt[NCH], lgs[NCH];
#pragma unroll
  for (int c = 0; c < NCH; ++c) {
    const float p = s_pred[c * 256 + t];
    const bool  m = (lab == c);
    mn[c]  = m ? p : __builtin_inff();
    cnt[c] = m ? 1.0f : 0.0f;
    lgs[c] = m ? lp : 0.0f;
  }

  // wave32 DPP butterfly reduction.
#pragma unroll
  for (int c = 0; c < NCH; ++c) wave_reduce3(mn[c], cnt[c], lgs[c]);

  const int wave = t >> 5;
  if ((t & 31) == 0) {
#pragma unroll
    for (int c = 0; c < NCH; ++c) {
      s_red[wave * 12 + c * 3 + 0] = mn[c];
      s_red[wave * 12 + c * 3 + 1] = cnt[c];
      s_red[wave * 12 + c * 3 + 2] = lgs[c];
    }
  }
  __syncthreads();

  // Cross-wave combine: 12 threads, one (channel,stat) slot each.
  if (t < 12) {
    const bool isMin = ((t % 3) == 0);
    float v = isMin ? __builtin_inff() : 0.0f;
#pragma unroll
    for (int w = 0; w < 8; ++w) {
      const float x = s_red[w * 12 + t];
      v = isMin ? fminf(v, x) : (v + x);
    }
    s_fin[t] = v;
  }
  __syncthreads();

  if (t == 0) {
    float acc = 0.0f;
#pragma unroll
    for (int c = 0; c < NCH; ++c) {
      const float tmn = s_fin[c * 3 + 0];
      const float tct = s_fin[c * 3 + 1];
      const float tls = s_fin[c * 3 + 2];
      const float minp = __builtin_isinf(tmn) ? 0.0f : tmn;  // posinf -> 0
      acc += (1.0f - minp) * tls / (tct + 1e-8f);            // empty -> 0
    }
    partial[(n * OH + py) * OW + px] = acc;   // one store, no global atomics
  }
}

// ---- final reduction: one block, no single-address atomic hot-spot --------
__global__ void __launch_bounds__(1024)
final_reduce_kernel(const float* __restrict__ partial,
                    float*       __restrict__ out,
                    int n, float scale) {
  __shared__ float s[32];
  float acc = 0.0f;
  for (int i = threadIdx.x; i < n; i += 1024) acc += partial[i];
#pragma unroll
  for (int m = 16; m >= 1; m >>= 1) acc += __shfl_xor(acc, m, 32);
  if ((threadIdx.x & 31) == 0) s[threadIdx.x >> 5] = acc;
  __syncthreads();
  if (threadIdx.x < 32) {
    float v = s[threadIdx.x];   // exactly 32 waves -> all slots valid
#pragma unroll
    for (int m = 16; m >= 1; m >>= 1) v += __shfl_xor(v, m, 32);
    if (threadIdx.x == 0) out[0] = -v * scale;
  }
}

extern "C" void kernel_launch(void* const* d_in, const int* in_sizes, int n_in,
                              void* d_out, int out_size, void* d_ws, size_t ws_size,
                              hipStream_t stream) {
  const float* preds  = (const float*)d_in[0];
  const int*   labels = (const int*)d_in[1];
  float*       out    = (float*)d_out;
  float*       part   = (float*)d_ws;   // 63504 floats = 248 KB scratch

  dim3 grid(OW, OH, NBATCH);
  patch_loss_kernel<<<grid, 256, 0, stream>>>(preds, labels, part);

  const int   nPart = NBATCH * OH * OW;                       // 63504
  const float scale = 1.0f / (float)(NBATCH * NCH * OH * OW); // 1/254016
  final_reduce_kernel<<<1, 1024, 0, stream>>>(part, out, nPart, scale);
}